// GAT_25288767439336
// MI455X (gfx1250) — compile-verified
//
#include <hip/hip_runtime.h>
#include <hip/hip_bf16.h>

typedef float v2f __attribute__((ext_vector_type(2)));
typedef float v8f __attribute__((ext_vector_type(8)));

#define NEG_SLOPE 0.2f
#define BN_EPS 1e-5f
// ordinal encoding of -inf for float atomic-max-via-int trick
#define NEG_ORD 0x807FFFFF

// ---------------------------------------------------------------------------
// float <-> monotone int ordinal (involution), enables atomicMax on floats
// ---------------------------------------------------------------------------
__device__ __forceinline__ int f2ord(float f) {
  int i = __float_as_int(f);
  return (i >= 0) ? i : (i ^ 0x7FFFFFFF);
}
__device__ __forceinline__ float ord2f(int i) {
  return __int_as_float((i >= 0) ? i : (i ^ 0x7FFFFFFF));
}

// ---------------------------------------------------------------------------
// GEMM: OUT[N,Fout] = X[N,Fin] @ W[Fin,Fout], fp32 WMMA 16x16x4.
// One wave computes a 16 x (16*TN) output slab; A fragment reused TN times.
// fp32 WMMA A-frag layout (16x4): lane l -> row (l&15), VGPR{0,1} hold
// K = 2*(l>>4) + {0,1}.  B-frag (4x16): lane l -> col (l&15), K = 2*(l>>4)+{0,1}.
// C/D (16x16): lane l -> col (l&15), VGPR i -> row i + 8*(l>>4).
// ---------------------------------------------------------------------------
template <int TN>
__global__ void gemm_wmma_f32(const float* __restrict__ X,
                              const float* __restrict__ W,
                              float* __restrict__ OUT,
                              int N, int Fin, int Fout) {
  const int lane = threadIdx.x & 31;
  const int wave = (int)((blockIdx.x * blockDim.x + threadIdx.x) >> 5);
  const int colGroups = Fout / (16 * TN);
  const int rowTile = wave / colGroups;
  const int cg = wave - rowTile * colGroups;
  if (rowTile * 16 >= N) return;  // whole wave exits together (EXEC stays full)

  const int mrow = rowTile * 16 + (lane & 15);
  const int khalf = lane >> 4;     // 0: K lanes 0..1, 1: K lanes 2..3
  const int ncol = lane & 15;

  v8f acc[TN];
#pragma unroll
  for (int t = 0; t < TN; ++t) acc[t] = (v8f){};

  const float* xrow = X + (size_t)mrow * Fin;
  for (int k = 0; k < Fin; k += 4) {
    v2f a = *(const v2f*)(xrow + k + 2 * khalf);
#pragma unroll
    for (int t = 0; t < TN; ++t) {
      const int col = (cg * TN + t) * 16 + ncol;
      v2f b;
      b.x = W[(size_t)(k + 2 * khalf) * Fout + col];
      b.y = W[(size_t)(k + 2 * khalf + 1) * Fout + col];
      acc[t] = __builtin_amdgcn_wmma_f32_16x16x4_f32(
          false, a, false, b, (short)0, acc[t], false, false);
    }
  }

#pragma unroll
  for (int t = 0; t < TN; ++t) {
    const int col = (cg * TN + t) * 16 + ncol;
#pragma unroll
    for (int i = 0; i < 8; ++i) {
      const int m = rowTile * 16 + khalf * 8 + i;
      OUT[(size_t)m * Fout + col] = acc[t][i];
    }
  }
}

// ---------------------------------------------------------------------------
// a_s[n,h] = dot(h[n,h,:], att_src[h,:]); a_d likewise. One wave per (n,h).
// ---------------------------------------------------------------------------
__global__ void attn_scores(const float* __restrict__ h,
                            const float* __restrict__ att_src,
                            const float* __restrict__ att_dst,
                            float* __restrict__ a_s, float* __restrict__ a_d,
                            int N, int H, int C) {
  const int wid = (int)((blockIdx.x * blockDim.x + threadIdx.x) >> 5);
  const int lane = threadIdx.x & 31;
  if (wid >= N * H) return;
  const int hh = wid % H;
  const float* hp = h + (size_t)wid * C;
  const float* sp = att_src + (size_t)hh * C;
  const float* dp = att_dst + (size_t)hh * C;
  float ss = 0.f, sd = 0.f;
  for (int c = lane; c < C; c += 32) {
    const float v = hp[c];
    ss += v * sp[c];
    sd += v * dp[c];
  }
#pragma unroll
  for (int off = 16; off > 0; off >>= 1) {
    ss += __shfl_xor(ss, off, 32);
    sd += __shfl_xor(sd, off, 32);
  }
  if (lane == 0) { a_s[wid] = ss; a_d[wid] = sd; }
}

__global__ void fill_i32(int* __restrict__ p, int v, long long n) {
  const long long t = (long long)blockIdx.x * blockDim.x + threadIdx.x;
  if (t < n) p[t] = v;
}

// ---------------------------------------------------------------------------
// Edge pass 1: m[dst,h] = max over incoming edges of leakyrelu(a_s[src]+a_d[dst])
// ---------------------------------------------------------------------------
__global__ void edge_max(const int* __restrict__ ei, int E, int N, int H,
                         const float* __restrict__ a_s,
                         const float* __restrict__ a_d,
                         int* __restrict__ m_ord) {
  const long long t = (long long)blockIdx.x * blockDim.x + threadIdx.x;
  const long long tot = (long long)(E + N) * H;
  if (t >= tot) return;
  const int e = (int)(t / H);
  const int hh = (int)(t - (long long)e * H);
  const int sn = (e < E) ? ei[e] : (e - E);
  const int dn = (e < E) ? ei[E + e] : (e - E);
  float ev = a_s[sn * H + hh] + a_d[dn * H + hh];
  ev = (ev > 0.f) ? ev : ev * NEG_SLOPE;
  atomicMax(&m_ord[dn * H + hh], f2ord(ev));
}

// ---------------------------------------------------------------------------
// Edge pass 2 (wave per (edge,head)): ex = exp(e - m[dst]);
//   s[dst,h] += ex;  out[dst,h,:] += ex * h[src,h,:]   (unnormalized)
// ---------------------------------------------------------------------------
__global__ void edge_accum(const int* __restrict__ ei, int E, int N, int H, int C,
                           const float* __restrict__ a_s,
                           const float* __restrict__ a_d,
                           const int* __restrict__ m_ord,
                           const float* __restrict__ h,
                           float* __restrict__ s_sum,
                           float* __restrict__ out) {
  const long long w = (long long)((blockIdx.x * blockDim.x + threadIdx.x) >> 5);
  const int lane = threadIdx.x & 31;
  const long long tot = (long long)(E + N) * H;
  if (w >= tot) return;
  const int e = (int)(w / H);
  const int hh = (int)(w - (long long)e * H);
  const int sn = (e < E) ? ei[e] : (e - E);
  const int dn = (e < E) ? ei[E + e] : (e - E);
  float ev = a_s[sn * H + hh] + a_d[dn * H + hh];
  ev = (ev > 0.f) ? ev : ev * NEG_SLOPE;
  const float mx = ord2f(m_ord[dn * H + hh]);
  const float ex = __expf(ev - mx);
  if (lane == 0) atomicAdd(&s_sum[dn * H + hh], ex);
  const float* hs = h + ((size_t)sn * H + hh) * C;
  float* od = out + ((size_t)dn * H + hh) * C;
  for (int c = lane; c < C; c += 32) atomicAdd(&od[c], ex * hs[c]);
}

// out[n,h,c] = out[n,h,c] / s[n,h] + bias[h*C+c]
__global__ void epilogue_div_bias(float* __restrict__ out,
                                  const float* __restrict__ s_sum,
                                  const float* __restrict__ bias,
                                  int N, int H, int C) {
  const long long t = (long long)blockIdx.x * blockDim.x + threadIdx.x;
  const long long tot = (long long)N * H * C;
  if (t >= tot) return;
  const int c = (int)(t % C);
  const long long nh = t / C;
  const int hh = (int)(nh % H);
  out[t] = out[t] / s_sum[nh] + bias[hh * C + c];
}

// block-per-column mean / rsqrt(var+eps)
__global__ void colstats(const float* __restrict__ x, float* __restrict__ mu,
                         float* __restrict__ rs, int N, int F) {
  const int f = blockIdx.x;
  __shared__ float sh1[256];
  __shared__ float sh2[256];
  float s1 = 0.f, s2 = 0.f;
  for (int r = threadIdx.x; r < N; r += blockDim.x) {
    const float v = x[(size_t)r * F + f];
    s1 += v;
    s2 += v * v;
  }
  sh1[threadIdx.x] = s1;
  sh2[threadIdx.x] = s2;
  __syncthreads();
  for (int off = 128; off > 0; off >>= 1) {
    if (threadIdx.x < off) {
      sh1[threadIdx.x] += sh1[threadIdx.x + off];
      sh2[threadIdx.x] += sh2[threadIdx.x + off];
    }
    __syncthreads();
  }
  if (threadIdx.x == 0) {
    const float m = sh1[0] / (float)N;
    const float var = sh2[0] / (float)N - m * m;
    mu[f] = m;
    rs[f] = rsqrtf(var + BN_EPS);
  }
}

__global__ void bn_elu(float* __restrict__ x, const float* __restrict__ mu,
                       const float* __restrict__ rs, const float* __restrict__ g,
                       const float* __restrict__ be, int N, int F) {
  const long long t = (long long)blockIdx.x * blockDim.x + threadIdx.x;
  const long long tot = (long long)N * F;
  if (t >= tot) return;
  const int f = (int)(t % F);
  const float v = (x[t] - mu[f]) * rs[f] * g[f] + be[f];
  x[t] = (v > 0.f) ? v : (__expf(v) - 1.f);
}

__global__ void logsoftmax_row(const float* __restrict__ x, float* __restrict__ out,
                               int N, int F) {
  const int n = blockIdx.x * blockDim.x + threadIdx.x;
  if (n >= N) return;
  const float* r = x + (size_t)n * F;
  float mx = r[0];
  for (int i = 1; i < F; ++i) mx = fmaxf(mx, r[i]);
  float s = 0.f;
  for (int i = 0; i < F; ++i) s += __expf(r[i] - mx);
  const float ls = __logf(s);
  float* o = out + (size_t)n * F;
  for (int i = 0; i < F; ++i) o[i] = r[i] - mx - ls;
}

// ---------------------------------------------------------------------------
// Host orchestration
// ---------------------------------------------------------------------------
struct LCfg { int Fin, H, C, TN, iW, iAs, iAd, iB, iG, iBe; };

extern "C" void kernel_launch(void* const* d_in, const int* in_sizes, int n_in,
                              void* d_out, int out_size, void* d_ws, size_t ws_size,
                              hipStream_t stream) {
  const int N = in_sizes[0] / 512;
  const int E = in_sizes[1] / 2;
  const float* x0 = (const float*)d_in[0];
  const int* ei = (const int*)d_in[1];

  static const LCfg cfgs[4] = {
      {512, 8, 64, 4, 2, 3, 4, 5, 6, 7},
      {512, 4, 64, 4, 8, 9, 10, 11, 12, 13},
      {256, 2, 64, 4, 14, 15, 16, 17, 18, 19},
      {128, 1, 16, 1, 20, 21, 22, 23, -1, -1},
  };

  // workspace carve-up (floats)
  float* hB    = (float*)d_ws;                   // [N,512] gemm output h
  float* outA  = hB + (size_t)N * 512;           // [N,512] aggregation / next input
  float* a_s   = outA + (size_t)N * 512;         // [N,8]
  float* a_d   = a_s + (size_t)N * 8;            // [N,8]
  int*   m_ord = (int*)(a_d + (size_t)N * 8);    // [N,8]
  float* s_sum = (float*)(m_ord + (size_t)N * 8);// [N,8]
  float* mu    = s_sum + (size_t)N * 8;          // [512]
  float* rs    = mu + 512;                       // [512]

  const float* X = x0;
  const int TPB = 256;

  for (int li = 0; li < 4; ++li) {
    const LCfg& c = cfgs[li];
    const int H = c.H, C = c.C, F = H * C, Fin = c.Fin;
    const float* W  = (const float*)d_in[c.iW];
    const float* as = (const float*)d_in[c.iAs];
    const float* ad = (const float*)d_in[c.iAd];
    const float* bi = (const float*)d_in[c.iB];

    // 1) h = X @ W (fp32 WMMA)
    {
      const long long waves = (long long)(N / 16) * (F / (16 * c.TN));
      const long long blocks = (waves * 32 + TPB - 1) / TPB;
      if (c.TN == 4)
        gemm_wmma_f32<4><<<(unsigned)blocks, TPB, 0, stream>>>(X, W, hB, N, Fin, F);
      else
        gemm_wmma_f32<1><<<(unsigned)blocks, TPB, 0, stream>>>(X, W, hB, N, Fin, F);
    }

    // 2) attention scores
    {
      const long long waves = (long long)N * H;
      const long long blocks = (waves * 32 + TPB - 1) / TPB;
      attn_scores<<<(unsigned)blocks, TPB, 0, stream>>>(hB, as, ad, a_s, a_d, N, H, C);
    }

    // 3) init accumulators (outA is safe to clear now: GEMM consumed it)
    {
      const long long nh = (long long)N * H;
      fill_i32<<<(unsigned)((nh + TPB - 1) / TPB), TPB, 0, stream>>>(m_ord, NEG_ORD, nh);
      hipMemsetAsync(s_sum, 0, (size_t)nh * sizeof(float), stream);
      hipMemsetAsync(outA, 0, (size_t)N * F * sizeof(float), stream);
    }

    // 4) segment max
    {
      const long long tot = (long long)(E + N) * H;
      edge_max<<<(unsigned)((tot + TPB - 1) / TPB), TPB, 0, stream>>>(
          ei, E, N, H, a_s, a_d, m_ord);
    }

    // 5) exp + weighted aggregation (unnormalized)
    {
      const long long tot = (long long)(E + N) * H;
      const long long blocks = (tot * 32 + TPB - 1) / TPB;
      edge_accum<<<(unsigned)blocks, TPB, 0, stream>>>(
          ei, E, N, H, C, a_s, a_d, m_ord, hB, s_sum, outA);
    }

    // 6) normalize + bias
    {
      const long long tot = (long long)N * F;
      epilogue_div_bias<<<(unsigned)((tot + TPB - 1) / TPB), TPB, 0, stream>>>(
          outA, s_sum, bi, N, H, C);
    }

    if (li < 3) {
      const float* g  = (const float*)d_in[c.iG];
      const float* be = (const float*)d_in[c.iBe];
      colstats<<<F, TPB, 0, stream>>>(outA, mu, rs, N, F);
      const long long tot = (long long)N * F;
      bn_elu<<<(unsigned)((tot + TPB - 1) / TPB), TPB, 0, stream>>>(
          outA, mu, rs, g, be, N, F);
      X = outA;
    } else {
      logsoftmax_row<<<(unsigned)((N + TPB - 1) / TPB), TPB, 0, stream>>>(
          outA, (float*)d_out, N, F);
    }
  }
}